// series_decomp_58858231824553
// MI455X (gfx1250) — compile-verified
//
#include <hip/hip_runtime.h>
#include <math.h>

// ---------------------------------------------------------------------------
// Problem constants (from the reference)
// ---------------------------------------------------------------------------
#define DIM_B 32
#define DIM_T 720
#define DIM_N 321
#define SIG_L (DIM_B * DIM_T * DIM_N)          // 7,395,840
#define MA_K  25
#define MA_P  12
#define THR_A 0.85f

static constexpr int L0C = SIG_L;              // 7,395,840
static constexpr int L1C = (L0C + 15) / 2;     // 3,697,927
static constexpr int L2C = (L1C + 15) / 2;     // 1,848,971
static constexpr int L3C = (L2C + 15) / 2;     //   924,493

// Buffer slot sizes padded to 16-float (64B) multiples so every coefficient
// array is 16B-aligned for float4/b128 access (rec buffers need +1 anyway).
static constexpr int padf(int L) { return (L + 16 + 15) & ~15; }
static constexpr int P1 = padf(L1C);
static constexpr int P2 = padf(L2C);
static constexpr int P3 = padf(L3C);

typedef float v2f __attribute__((ext_vector_type(2)));
typedef float v8f __attribute__((ext_vector_type(8)));

// Flipped decomposition filters: FLO[k] = DEC_LO[15-k], FHI[k] = DEC_HI[15-k].
// Note REC_LO == FLO and REC_HI == FHI (reconstruction filters are the
// reversed decomposition filters), so these two tables serve both directions.
__constant__ float FLO[16] = {
     0.0018899503327594609f, -0.0003029205147213668f, -0.0149522583370482300f,
     0.0038087520138906150f,  0.0491371796736075060f, -0.0272190299170560030f,
    -0.0519458381077090400f,  0.3644418948353389500f,  0.7771857517005235000f,
     0.4813596512583722000f, -0.0612733590676585240f, -0.1432942383508097000f,
     0.0076074873249176050f,  0.0316950878114929800f, -0.0005421323317911481f,
    -0.0033824159510061256f
};
__constant__ float FHI[16] = {
    -0.0033824159510061256f,  0.0005421323317911481f,  0.0316950878114929800f,
    -0.0076074873249176050f, -0.1432942383508097000f,  0.0612733590676585240f,
     0.4813596512583722000f, -0.7771857517005235000f,  0.3644418948353389500f,
     0.0519458381077090400f, -0.0272190299170560030f, -0.0491371796736075060f,
     0.0038087520138906150f,  0.0149522583370482300f, -0.0003029205147213668f,
    -0.0018899503327594609f
};

// ---------------------------------------------------------------------------
// 1) Moving average along T with edge replication; also emits residual.
// ---------------------------------------------------------------------------
__global__ __launch_bounds__(256) void movavg_kernel(const float* __restrict__ x,
                                                     float* __restrict__ trend,
                                                     float* __restrict__ res) {
    int i = blockIdx.x * 256 + threadIdx.x;
    if (i >= L0C) return;
    int n  = i % DIM_N;
    int bt = i / DIM_N;
    int t  = bt % DIM_T;
    int b  = bt / DIM_T;
    const float* col = x + (size_t)b * DIM_T * DIM_N + n;
    float s = 0.0f;
#pragma unroll
    for (int j = -MA_P; j <= MA_P; ++j) {
        int tc = t + j;
        tc = tc < 0 ? 0 : (tc >= DIM_T ? DIM_T - 1 : tc);
        s += col[(size_t)tc * DIM_N];
    }
    float tr = s / 25.0f;
    trend[i] = tr;
    res[i]   = x[i] - tr;
}

// ---------------------------------------------------------------------------
// 2) Forward DWT level via V_WMMA_F32_16X16X4_F32.
//    C(16x16) = F(16x32) x S(32x16), chained as 8 WMMAs with K=4.
//    A rows 2j/2j+1 hold lo/hi filters shifted by 2j taps; B columns are
//    stride-16 im2col slices of the symmetric-extended signal (from LDS).
//    One wave -> 128 ca + 128 cd contiguous outputs.  The C tile is scattered
//    through LDS so the global write-back is fully-coalesced float4 stores
//    (block-uniform tail guard: only the final block takes the slow path).
// ---------------------------------------------------------------------------
__global__ __launch_bounds__(256) void dwt_wmma_kernel(const float* __restrict__ a, int La,
                                                       float* __restrict__ ca,
                                                       float* __restrict__ cd, int Lo) {
    __shared__ __align__(16) float sin_[2064];   // input window (8 waves * 256 + 16)
    __shared__ __align__(16) float sout[2048];   // [0..1023]=ca band, [1024..2047]=cd band
    const int tid   = threadIdx.x;
    const int wave  = tid >> 5;
    const int lane  = tid & 31;
    const int gBase = blockIdx.x * 1024;         // outputs per block
    const int g0    = 2 * gBase - 14;            // ext-window start in a[]

    // Coalesced LDS fill with symmetric ('symmetric' mode) reflection.
    for (int i = tid; i < 2064; i += 256) {
        int q = g0 + i;
        q = (q < 0)   ? (-q - 1)         : q;
        q = (q >= La) ? (2 * La - 1 - q) : q;
        q = (q < 0)   ? 0 : ((q >= La) ? La - 1 : q);
        sin_[i] = a[q];
    }
    __syncthreads();

    // A fragment (16x4 per step): lanes 0-15 hold K={k0,k0+1}, lanes 16-31 K={k0+2,k0+3}.
    const int m   = lane & 15;          // A row / B column index (same expression)
    const int kh  = (lane >> 4) << 1;   // 0 or 2
    const int shf = m & ~1;             // filter shift (2j)
    const int fil = m & 1;              // 0 = lo, 1 = hi
    v2f Afrag[8];
#pragma unroll
    for (int c = 0; c < 8; ++c) {
        int k0 = (c << 2) + kh;
        int i0 = k0 - shf;
        int i1 = i0 + 1;
        Afrag[c].x = (i0 >= 0 && i0 < 16) ? (fil ? FHI[i0] : FLO[i0]) : 0.0f;
        Afrag[c].y = (i1 >= 0 && i1 < 16) ? (fil ? FHI[i1] : FLO[i1]) : 0.0f;
    }

    // B fragment: B[k][n] = sin_[256*wave + 16*n + k]  (k = 4c + kh + {0,1})
    const int sbase = (wave << 8) + (m << 4) + kh;
    v8f acc = {};
#pragma unroll
    for (int c = 0; c < 8; ++c) {
        v2f Bfrag;
        Bfrag.x = sin_[sbase + (c << 2)];
        Bfrag.y = sin_[sbase + (c << 2) + 1];
        acc = __builtin_amdgcn_wmma_f32_16x16x4_f32(false, Afrag[c], false, Bfrag,
                                                    (short)0, acc, false, false);
    }

    // C layout: VGPR v, lanes 0-15 -> row v, lanes 16-31 -> row v+8; col = lane&15.
    // Row 2j+f of column n is block-local output (128*wave + 8n + j), band f.
    const int mh = (lane >> 4) << 3;
#pragma unroll
    for (int v = 0; v < 8; ++v) {
        int mm     = v + mh;
        int oLocal = (wave << 7) + (m << 3) + (mm >> 1);
        sout[((mm & 1) << 10) + oLocal] = acc[v];
    }
    __syncthreads();

    // Coalesced write-back: one float4 per thread per band.
    const int o = gBase + (tid << 2);
    if (gBase + 1024 <= Lo) {                    // block-uniform fast path
        *reinterpret_cast<float4*>(ca + o) =
            *reinterpret_cast<const float4*>(&sout[tid << 2]);
        *reinterpret_cast<float4*>(cd + o) =
            *reinterpret_cast<const float4*>(&sout[1024 + (tid << 2)]);
    } else {
#pragma unroll
        for (int e = 0; e < 4; ++e) {
            if (o + e < Lo) {
                ca[o + e] = sout[(tid << 2) + e];
                cd[o + e] = sout[1024 + (tid << 2) + e];
            }
        }
    }
}

// ---------------------------------------------------------------------------
// 3) Median of |cd1| via two-pass radix selection on the uint ordering of
//    non-negative float bits (exact middle element; deterministic atomics).
// ---------------------------------------------------------------------------
__global__ __launch_bounds__(256) void zero_kernel(unsigned* __restrict__ p, int n) {
    int i = blockIdx.x * 256 + threadIdx.x;
    if (i < n) p[i] = 0u;
}

__global__ __launch_bounds__(256) void hist1_kernel(const float* __restrict__ d, int n,
                                                    unsigned* __restrict__ hist) {
    int i = blockIdx.x * 256 + threadIdx.x;
    if (i >= n) return;
    unsigned key = __float_as_uint(d[i]) & 0x7FFFFFFFu;
    atomicAdd(&hist[key >> 15], 1u);
}

__global__ __launch_bounds__(256) void select1_kernel(const unsigned* __restrict__ hist,
                                                      unsigned rank,
                                                      unsigned* __restrict__ params) {
    __shared__ unsigned psum[256];
    int t = threadIdx.x;
    unsigned ssum = 0;
    for (int i = 0; i < 256; ++i) ssum += hist[t * 256 + i];
    psum[t] = ssum;
    __syncthreads();
    if (t == 0) {
        unsigned cum = 0;
        int chunk = 255;
        for (int i = 0; i < 256; ++i) {
            if (cum + psum[i] > rank) { chunk = i; break; }
            cum += psum[i];
        }
        int bin = chunk * 256 + 255;
        unsigned c2 = cum;
        for (int i = 0; i < 256; ++i) {
            unsigned h = hist[chunk * 256 + i];
            if (c2 + h > rank) { bin = chunk * 256 + i; break; }
            c2 += h;
        }
        params[1] = (unsigned)bin;     // high 16 bits of median key
        params[2] = rank - c2;         // remaining rank inside that bin
    }
}

__global__ __launch_bounds__(256) void hist2_kernel(const float* __restrict__ d, int n,
                                                    unsigned* __restrict__ hist2,
                                                    const unsigned* __restrict__ params) {
    int i = blockIdx.x * 256 + threadIdx.x;
    if (i >= n) return;
    unsigned key = __float_as_uint(d[i]) & 0x7FFFFFFFu;
    if ((key >> 15) == params[1]) atomicAdd(&hist2[key & 0x7FFFu], 1u);
}

__global__ __launch_bounds__(256) void select2_kernel(const unsigned* __restrict__ hist2,
                                                      unsigned* __restrict__ params,
                                                      int Lsig) {
    __shared__ unsigned psum[256];
    int t = threadIdx.x;
    unsigned s = 0;
    for (int i = 0; i < 128; ++i) s += hist2[t * 128 + i];
    psum[t] = s;
    __syncthreads();
    if (t == 0) {
        unsigned rank = params[2];
        unsigned B1   = params[1];
        unsigned cum  = 0;
        int chunk = 255;
        for (int i = 0; i < 256; ++i) {
            if (cum + psum[i] > rank) { chunk = i; break; }
            cum += psum[i];
        }
        int bin2 = chunk * 128 + 127;
        for (int i = 0; i < 128; ++i) {
            unsigned h = hist2[chunk * 128 + i];
            if (cum + h > rank) { bin2 = chunk * 128 + i; break; }
            cum += h;
        }
        unsigned key = (B1 << 15) | (unsigned)bin2;
        float med   = __uint_as_float(key);
        float sigma = med / 0.6745f;
        float lam   = sigma * sqrtf(2.0f * logf((float)Lsig));
        ((float*)params)[4] = lam;
        ((float*)params)[5] = THR_A * lam;
    }
}

// ---------------------------------------------------------------------------
// 4) Soft-ish threshold (in place): |d|>=lam ? sign(d)*(|d|-A*lam) : 0
// ---------------------------------------------------------------------------
__global__ __launch_bounds__(256) void thresh_kernel(float* __restrict__ d, int n,
                                                     const unsigned* __restrict__ params) {
    int i = blockIdx.x * 256 + threadIdx.x;
    if (i >= n) return;
    float lam  = ((const float*)params)[4];
    float lamA = ((const float*)params)[5];
    float v  = d[i];
    float av = fabsf(v);
    d[i] = (av >= lam) ? copysignf(av - lamA, v) : 0.0f;
}

// ---------------------------------------------------------------------------
// 5) Inverse DWT: out[i] = sum_p ca[p]*REC_LO[i+14-2p] + cd[p]*REC_HI[i+14-2p]
//    (REC_LO == FLO, REC_HI == FHI).  Lout = 2*Lc - 14.
// ---------------------------------------------------------------------------
__global__ __launch_bounds__(256) void idwt_kernel(const float* __restrict__ ca,
                                                   const float* __restrict__ cd,
                                                   int Lc, float* __restrict__ out, int Lout) {
    int i = blockIdx.x * 256 + threadIdx.x;
    if (i >= Lout) return;
    int pLo = i >> 1;
    int pHi = (i + 14) >> 1;
    if (pHi > Lc - 1) pHi = Lc - 1;
    float acc = 0.0f;
    for (int p = pLo; p <= pHi; ++p) {
        int j = i + 14 - 2 * p;       // in [0,16)
        acc = fmaf(ca[p], FLO[j], acc);
        acc = fmaf(cd[p], FHI[j], acc);
    }
    out[i] = acc;
}

// ---------------------------------------------------------------------------
// 6) diff: out[i] = base[i] - nn[i]  (out may alias base)
// ---------------------------------------------------------------------------
__global__ __launch_bounds__(256) void diff_kernel(float* __restrict__ out,
                                                   const float* __restrict__ nn, int n) {
    int i = blockIdx.x * 256 + threadIdx.x;
    if (i >= n) return;
    out[i] = out[i] - nn[i];
}

// ---------------------------------------------------------------------------
// Host-side pipeline
// ---------------------------------------------------------------------------
static inline int cdiv(int a, int b) { return (a + b - 1) / b; }

static void launch_dwt(const float* a, int La, float* ca, float* cd, int Lo,
                       hipStream_t stream) {
    int blocks = cdiv(Lo, 1024);
    dwt_wmma_kernel<<<blocks, 256, 0, stream>>>(a, La, ca, cd, Lo);
}

static void denoise_pipeline(const float* sig, float* nn,
                             float* ca1, float* cd1, float* ca2, float* cd2,
                             float* ca3, float* cd3,
                             unsigned* hist1, unsigned* hist2, unsigned* params,
                             hipStream_t stream) {
    // Forward DWT, 3 levels (WMMA path)
    launch_dwt(sig, L0C, ca1, cd1, L1C, stream);
    launch_dwt(ca1, L1C, ca2, cd2, L2C, stream);
    launch_dwt(ca2, L2C, ca3, cd3, L3C, stream);

    // lam = (median(|cd1|)/0.6745) * sqrt(2 ln L0)
    const int nhist = 65536 + 32768 + 16;
    zero_kernel<<<cdiv(nhist, 256), 256, 0, stream>>>(hist1, nhist);
    hist1_kernel<<<cdiv(L1C, 256), 256, 0, stream>>>(cd1, L1C, hist1);
    select1_kernel<<<1, 256, 0, stream>>>(hist1, (unsigned)((L1C - 1) / 2), params);
    hist2_kernel<<<cdiv(L1C, 256), 256, 0, stream>>>(cd1, L1C, hist2, params);
    select2_kernel<<<1, 256, 0, stream>>>(hist2, params, L0C);

    // Threshold detail bands in place
    thresh_kernel<<<cdiv(L1C, 256), 256, 0, stream>>>(cd1, L1C, params);
    thresh_kernel<<<cdiv(L2C, 256), 256, 0, stream>>>(cd2, L2C, params);
    thresh_kernel<<<cdiv(L3C, 256), 256, 0, stream>>>(cd3, L3C, params);

    // Inverse DWT (rec2 reuses ca2's slot, rec1 reuses ca1's slot; the
    // "drop last sample of a" crop is realized by passing Lc = len(d)).
    int r2 = 2 * L3C - 14;   // == L2C + 1
    idwt_kernel<<<cdiv(r2, 256), 256, 0, stream>>>(ca3, cd3, L3C, ca2, r2);
    int r1 = 2 * L2C - 14;   // == L1C + 1
    idwt_kernel<<<cdiv(r1, 256), 256, 0, stream>>>(ca2, cd2, L2C, ca1, r1);
    idwt_kernel<<<cdiv(L0C, 256), 256, 0, stream>>>(ca1, cd1, L1C, nn, L0C);
}

extern "C" void kernel_launch(void* const* d_in, const int* in_sizes, int n_in,
                              void* d_out, int out_size, void* d_ws, size_t ws_size,
                              hipStream_t stream) {
    (void)in_sizes; (void)n_in; (void)out_size; (void)ws_size;
    const float* x = (const float*)d_in[0];
    float* out = (float*)d_out;

    // d_out slots double as scratch: slot0/slot2 hold trend/res until the
    // final diff converts them to (trend-trend_nn)/(res-res_nn) in place.
    float* o_tdiff = out;                       // trend, then trend - trend_nn
    float* o_tnn   = out + (size_t)L0C;         // trend_nn
    float* o_rdiff = out + (size_t)2 * L0C;     // res, then res - res_nn
    float* o_rnn   = out + (size_t)3 * L0C;     // res_nn

    // Workspace: coefficient buffers (ca slots reused as idwt rec buffers,
    // slot sizes padded to 64B so float4/b128 write-back stays aligned) +
    // histograms.  Total use: ~52.2 MB.
    float* ws  = (float*)d_ws;
    float* ca1 = ws;
    float* cd1 = ca1 + P1;
    float* ca2 = cd1 + P1;
    float* cd2 = ca2 + P2;
    float* ca3 = cd2 + P2;
    float* cd3 = ca3 + P3;
    unsigned* hist1  = (unsigned*)(cd3 + P3);
    unsigned* hist2  = hist1 + 65536;
    unsigned* params = hist2 + 32768;

    movavg_kernel<<<cdiv(L0C, 256), 256, 0, stream>>>(x, o_tdiff, o_rdiff);

    denoise_pipeline(o_tdiff, o_tnn, ca1, cd1, ca2, cd2, ca3, cd3,
                     hist1, hist2, params, stream);
    diff_kernel<<<cdiv(L0C, 256), 256, 0, stream>>>(o_tdiff, o_tnn, L0C);

    denoise_pipeline(o_rdiff, o_rnn, ca1, cd1, ca2, cd2, ca3, cd3,
                     hist1, hist2, params, stream);
    diff_kernel<<<cdiv(L0C, 256), 256, 0, stream>>>(o_rdiff, o_rnn, L0C);
}